// BlockSparseLinearWithPerm_44427141710516
// MI455X (gfx1250) — compile-verified
//
#include <hip/hip_runtime.h>

typedef float v2f __attribute__((ext_vector_type(2)));
typedef float v4f __attribute__((ext_vector_type(4)));
typedef float v8f __attribute__((ext_vector_type(8)));
typedef int   v4i __attribute__((ext_vector_type(4)));

#define D_IN     1024
#define D_OUT    1024
#define BS       64
#define NNZ      64
#define NBR      16          // D_OUT / BS block rows
#define ROWS     16          // x rows per workgroup
#define XSTR     (D_IN + 4)  // LDS row stride (floats): bank-conflict-free b64 column access
#define YSTR     (D_OUT + 4)
#define NTHREADS 128         // 4 wave32

// ---------- setup kernel: counting-sort blocks by brow, pack (k<<8)|(br<<4)|(bc) ----------
__global__ void build_order_kernel(const int* __restrict__ brow,
                                   const int* __restrict__ bcol,
                                   int* __restrict__ ord) {
  if (threadIdx.x == 0 && blockIdx.x == 0) {
    int cnt[NBR];
    #pragma unroll
    for (int i = 0; i < NBR; ++i) cnt[i] = 0;
    for (int k = 0; k < NNZ; ++k) cnt[brow[k]]++;
    int pos[NBR]; int run = 0;
    #pragma unroll
    for (int i = 0; i < NBR; ++i) { pos[i] = run; run += cnt[i]; }
    for (int k = 0; k < NNZ; ++k) {
      int b = brow[k];
      ord[pos[b]++] = (k << 8) | (b << 4) | bcol[k];
    }
  }
}

extern __shared__ float smem[];

// D = A x B + C, fp32, 16x16x4
#define WMMA4(A_, B_, C_) \
  C_ = __builtin_amdgcn_wmma_f32_16x16x4_f32(false, A_, false, B_, (short)0, C_, false, false)

// issue 16 v2f global loads of the B fragment for block (PK>>8) into BUF (no wait here)
#define LOAD_B(BUF, PK) do {                                                     \
    const float* _wb = wblocks + (size_t)((PK) >> 8) * (BS * BS) + wrow_off;     \
    _Pragma("unroll") for (int _j = 0; _j < 16; ++_j)                            \
      BUF[_j] = *(const v2f*)(_wb + _j * 4);                                     \
  } while (0)

// 16 WMMA (4 independent accumulator chains), A from LDS just-in-time
#define COMPUTE(BUF, PK) do {                                                    \
    const float* _ax = lds_xg + arow_off + ((PK) & 15) * BS;                     \
    _Pragma("unroll") for (int _kk = 0; _kk < 4; ++_kk) {                        \
      v2f _a0 = *(const v2f*)(_ax + _kk * 16 + 0);  WMMA4(_a0, BUF[_kk*4+0], acc0); \
      v2f _a1 = *(const v2f*)(_ax + _kk * 16 + 4);  WMMA4(_a1, BUF[_kk*4+1], acc1); \
      v2f _a2 = *(const v2f*)(_ax + _kk * 16 + 8);  WMMA4(_a2, BUF[_kk*4+2], acc2); \
      v2f _a3 = *(const v2f*)(_ax + _kk * 16 + 12); WMMA4(_a3, BUF[_kk*4+3], acc3); \
    }                                                                            \
  } while (0)

// accumulate C tile onto bias-initialized y tile in LDS, then zero accumulators
#define FLUSH(CBR) do {                                                          \
    float* _yc = lds_y + (CBR) * BS + ycol_off;                                  \
    _Pragma("unroll") for (int _p = 0; _p < 8; ++_p)                             \
      _yc[(_p + 8 * half) * YSTR] += acc0[_p] + acc1[_p] + acc2[_p] + acc3[_p];  \
    _Pragma("unroll") for (int _p = 0; _p < 8; ++_p)                             \
      { acc0[_p] = 0.f; acc1[_p] = 0.f; acc2[_p] = 0.f; acc3[_p] = 0.f; }        \
  } while (0)

__global__ __launch_bounds__(NTHREADS)
void bsl_wmma_kernel(const float* __restrict__ x,
                     const int*   __restrict__ in_perm,
                     const int*   __restrict__ out_perm,
                     const float* __restrict__ wblocks,   // [NNZ][BS][BS]
                     const float* __restrict__ bias,
                     const int*   __restrict__ ord,       // brow-sorted packed block list
                     float*       __restrict__ out)
{
  float* lds_xg = smem;                      // ROWS * XSTR  gathered activations
  float* lds_y  = smem + ROWS * XSTR;        // ROWS * YSTR  y tile (also raw-x staging)

  const int  tid  = threadIdx.x;
  const long row0 = (long)blockIdx.x * ROWS;

  // ---- 1) coalesced load of raw x tile into lds_y (staging area) ----
  for (int i = tid; i < ROWS * (D_IN / 4); i += NTHREADS) {
    int r  = i / (D_IN / 4);
    int c4 = (i % (D_IN / 4)) * 4;
    v4f v = *(const v4f*)(x + (row0 + r) * D_IN + c4);
    lds_y[r * YSTR + c4 + 0] = v.x;
    lds_y[r * YSTR + c4 + 1] = v.y;
    lds_y[r * YSTR + c4 + 2] = v.z;
    lds_y[r * YSTR + c4 + 3] = v.w;
  }
  __syncthreads();

  // ---- 2) in-LDS column gather: xg[r][c] = raw[r][in_perm[c]] ----
  for (int i = tid; i < ROWS * D_IN; i += NTHREADS) {
    int r = i >> 10;
    int c = i & (D_IN - 1);
    lds_xg[r * XSTR + c] = lds_y[r * YSTR + in_perm[c]];
  }
  __syncthreads();

  // ---- 3) init y tile with bias (covers block-rows with no nnz blocks) ----
  for (int i = tid; i < ROWS * D_OUT; i += NTHREADS) {
    int r = i >> 10;
    int c = i & (D_OUT - 1);
    lds_y[r * YSTR + c] = bias[c];
  }
  __syncthreads();

  // ---- 4) block-sparse matmul: brow-sorted flat walk, B register double-buffer ----
  const int wave = tid >> 5;     // 0..3 : 16-wide output column slice
  const int lane = tid & 31;
  const int l15  = lane & 15;    // A: M row | B/C: N col
  const int half = lane >> 4;    // K phase for A/B; +8 M rows for C/D

  const int wrow_off = (wave * 16 + l15) * BS + 2 * half;  // W block row o, K phase
  const int arow_off = l15 * XSTR + 2 * half;              // xg row, K phase
  const int ycol_off = wave * 16 + l15;                    // y column within block row

  v2f bufX[16], bufY[16];
  v8f acc0 = {}, acc1 = {}, acc2 = {}, acc3 = {};

  int cur = __builtin_amdgcn_readfirstlane(ord[0]);
  LOAD_B(bufX, cur);

  for (int base = 0; base < NNZ; base += 2) {
    // --- iteration base: current block in bufX, prefetch next into bufY ---
    {
      int n1  = (base + 1 < NNZ) ? base + 1 : base;
      int pk1 = __builtin_amdgcn_readfirstlane(ord[n1]);
      LOAD_B(bufY, pk1);
      COMPUTE(bufX, cur);
      int cbr = (cur >> 4) & 15, nbr = (pk1 >> 4) & 15;
      if (base + 1 >= NNZ || nbr != cbr) FLUSH(cbr);
      cur = pk1;
    }
    // --- iteration base+1: current block in bufY, prefetch next into bufX ---
    {
      int n2  = (base + 2 < NNZ) ? base + 2 : NNZ - 1;
      int pk2 = __builtin_amdgcn_readfirstlane(ord[n2]);
      LOAD_B(bufX, pk2);
      COMPUTE(bufY, cur);
      int cbr = (cur >> 4) & 15, nbr = (pk2 >> 4) & 15;
      if (base + 2 >= NNZ || nbr != cbr) FLUSH(cbr);
      cur = pk2;
    }
  }
  __syncthreads();

  // ---- 5) out_perm gather from LDS, coalesced float4 stores ----
  for (int i = tid; i < ROWS * (D_OUT / 4); i += NTHREADS) {
    int r  = i / (D_OUT / 4);
    int o4 = (i % (D_OUT / 4)) * 4;
    v4i p = *(const v4i*)(out_perm + o4);
    v4f v;
    v.x = lds_y[r * YSTR + p.x];
    v.y = lds_y[r * YSTR + p.y];
    v.z = lds_y[r * YSTR + p.z];
    v.w = lds_y[r * YSTR + p.w];
    *(v4f*)(out + (row0 + r) * D_OUT + o4) = v;
  }
}

extern "C" void kernel_launch(void* const* d_in, const int* in_sizes, int n_in,
                              void* d_out, int out_size, void* d_ws, size_t ws_size,
                              hipStream_t stream) {
  (void)n_in; (void)out_size; (void)ws_size;
  const float* x        = (const float*)d_in[0];
  const int*   in_perm  = (const int*)  d_in[1];
  const int*   out_perm = (const int*)  d_in[2];
  const float* wblocks  = (const float*)d_in[3];
  const int*   brow     = (const int*)  d_in[4];
  const int*   bcol     = (const int*)  d_in[5];
  const float* bias     = (const float*)d_in[6];
  float*       out      = (float*)d_out;
  int*         ord      = (int*)d_ws;   // NNZ ints of scratch

  const int rows = in_sizes[0] / D_IN;  // 65536
  const int grid = rows / ROWS;         // 4096 workgroups

  build_order_kernel<<<1, 1, 0, stream>>>(brow, bcol, ord);

  const size_t shmem = (size_t)(ROWS * XSTR + ROWS * YSTR) * sizeof(float);  // ~131.6 KB
  hipFuncSetAttribute(reinterpret_cast<const void*>(bsl_wmma_kernel),
                      hipFuncAttributeMaxDynamicSharedMemorySize, (int)shmem);

  bsl_wmma_kernel<<<grid, NTHREADS, shmem, stream>>>(
      x, in_perm, out_perm, wblocks, bias, ord, out);
}